// DiversityRegularizer_74002286510639
// MI455X (gfx1250) — compile-verified
//
#include <hip/hip_runtime.h>

// Cosine-diagonal mean for [N,128] fp32 pairs, tuned for MI455X (gfx1250).
// Bandwidth-bound: 1 GiB read / 23.3 TB/s ~= 46 us floor. We stream tiles of
// 16 rows and use V_WMMA_F32_16X16X4_F32 to form diag(A*B^T), diag(A*A^T),
// diag(B*B^T) with ~3x fewer instructions per byte than scalar FMA+shuffle.

typedef float v2f __attribute__((ext_vector_type(2)));
typedef float v8f __attribute__((ext_vector_type(8)));

#define D_DIM 128
#define TILE_M 16
#define EPS_NORM 1e-12f

__device__ __forceinline__ float pick8(v8f c, int idx) {
  // Per-lane variable select of one accumulator component (cndmask chain).
  float r = c[0];
  r = (idx == 1) ? c[1] : r;
  r = (idx == 2) ? c[2] : r;
  r = (idx == 3) ? c[3] : r;
  r = (idx == 4) ? c[4] : r;
  r = (idx == 5) ? c[5] : r;
  r = (idx == 6) ? c[6] : r;
  r = (idx == 7) ? c[7] : r;
  return r;
}

__global__ void __launch_bounds__(256)
cosdiag_wmma_kernel(const float* __restrict__ a, const float* __restrict__ b,
                    float* __restrict__ partial, int nTiles) {
  const int lane = threadIdx.x & 31;
  const int wid  = threadIdx.x >> 5;
  const int wavesPerBlock = blockDim.x >> 5;
  const int gwave  = blockIdx.x * wavesPerBlock + wid;
  const int nWaves = gridDim.x * wavesPerBlock;

  // WMMA f32 16x16x4 operand layout: lane l supplies row (l&15),
  // columns k0 + 2*(l>>4) and k0 + 2*(l>>4)+1  ->  one float2 load per lane
  // serves as BOTH the A-operand (16x4) and B-operand (4x16) register image.
  const int mrow = lane & 15;
  const int koff = (lane >> 4) << 1;  // 0 or 2

  float acc = 0.0f;

  for (int t = gwave; t < nTiles; t += nWaves) {
    const size_t rowBase = (size_t)t * TILE_M + (size_t)mrow;
    const float* pa = a + rowBase * D_DIM + koff;
    const float* pb = b + rowBase * D_DIM + koff;

    // Prefetch next tile (uniform branch; EXEC stays all-ones for WMMA).
    if (t + nWaves < nTiles) {
      const size_t nb_ = (size_t)(t + nWaves) * TILE_M + (size_t)mrow;
      __builtin_prefetch(a + nb_ * D_DIM + koff, 0, 0);
      __builtin_prefetch(b + nb_ * D_DIM + koff, 0, 0);
    }

    v8f cab = {};  // diag -> a.b
    v8f caa = {};  // diag -> a.a
    v8f cbb = {};  // diag -> b.b
#pragma unroll
    for (int k0 = 0; k0 < D_DIM; k0 += 4) {
      v2f va = *(const v2f*)(pa + k0);
      v2f vb = *(const v2f*)(pb + k0);
      cab = __builtin_amdgcn_wmma_f32_16x16x4_f32(false, va, false, vb,
                                                  (short)0, cab, false, false);
      caa = __builtin_amdgcn_wmma_f32_16x16x4_f32(false, va, false, va,
                                                  (short)0, caa, false, false);
      cbb = __builtin_amdgcn_wmma_f32_16x16x4_f32(false, vb, false, vb,
                                                  (short)0, cbb, false, false);
    }

    // 16x16 f32 C/D layout: element (i,i) lives at lane i, comp i (i<8) and
    // lane i+16, comp i-8 (i>=8). So lanes 0..7 and 24..31 own the diagonal,
    // with component index (lane & 7).
    const int  idx   = lane & 7;
    const bool valid = (lane < 8) || (lane >= 24);
    float ab = pick8(cab, idx);
    float aa = pick8(caa, idx);
    float bb = pick8(cbb, idx);
    float na = fmaxf(__builtin_sqrtf(aa), EPS_NORM);
    float nb = fmaxf(__builtin_sqrtf(bb), EPS_NORM);
    float c  = ab / (na * nb);
    acc += valid ? c : 0.0f;
  }

  // Wave32 butterfly reduction.
  for (int off = 16; off > 0; off >>= 1)
    acc += __shfl_xor(acc, off, 32);

  __shared__ float smem[8];
  if (lane == 0) smem[wid] = acc;
  __syncthreads();
  if (threadIdx.x == 0) {
    float tsum = 0.0f;
    for (int i = 0; i < wavesPerBlock; ++i) tsum += smem[i];
    partial[blockIdx.x] = tsum;  // one deterministic store per block
  }
}

__global__ void __launch_bounds__(256)
cosdiag_finalize_kernel(const float* __restrict__ partial, int nPartials,
                        float* __restrict__ out, float invN) {
  __shared__ float smem[256];
  float t = 0.0f;
  for (int i = threadIdx.x; i < nPartials; i += blockDim.x) t += partial[i];
  smem[threadIdx.x] = t;
  __syncthreads();
  for (int stride = 128; stride > 0; stride >>= 1) {
    if ((int)threadIdx.x < stride) smem[threadIdx.x] += smem[threadIdx.x + stride];
    __syncthreads();
  }
  if (threadIdx.x == 0) out[0] = smem[0] * invN;
}

extern "C" void kernel_launch(void* const* d_in, const int* in_sizes, int n_in,
                              void* d_out, int out_size, void* d_ws, size_t ws_size,
                              hipStream_t stream) {
  const float* a = (const float*)d_in[0];
  const float* b = (const float*)d_in[1];
  float* out = (float*)d_out;

  const int N      = in_sizes[0] / D_DIM;   // 1048576 rows
  const int nTiles = N / TILE_M;            // 65536 tiles of 16 rows

  int blocks = 2048;                        // 16384 waves -> 4 tiles/wave
  if ((size_t)blocks * sizeof(float) > ws_size)
    blocks = (int)(ws_size / sizeof(float));
  if (blocks < 1) blocks = 1;

  float* partial = (float*)d_ws;
  cosdiag_wmma_kernel<<<blocks, 256, 0, stream>>>(a, b, partial, nTiles);
  cosdiag_finalize_kernel<<<1, 256, 0, stream>>>(partial, blocks, out,
                                                 1.0f / (float)N);
}